// NaiveGCN_24696061952298
// MI455X (gfx1250) — compile-verified
//
#include <hip/hip_runtime.h>
#include <math.h>

typedef __attribute__((ext_vector_type(2))) float v2f;
typedef __attribute__((ext_vector_type(8))) float v8f;

// ---------------- utility kernels ----------------

__global__ void zero_f32(float* __restrict__ p, long long n) {
    long long i = (long long)blockIdx.x * blockDim.x + threadIdx.x;
    long long stride = (long long)gridDim.x * blockDim.x;
    for (; i < n; i += stride) p[i] = 0.0f;
}

__global__ void degree_count(const int* __restrict__ dst, unsigned* __restrict__ cnt, int E) {
    int e = blockIdx.x * blockDim.x + threadIdx.x;
    if (e < E) atomicAdd(&cnt[dst[e]], 1u);
}

__global__ void inv_degree(const unsigned* __restrict__ cnt, float* __restrict__ inv, int n) {
    int i = blockIdx.x * blockDim.x + threadIdx.x;
    if (i < n) {
        unsigned c = cnt[i];
        inv[i] = 1.0f / (float)(c ? c : 1u);
    }
}

// one thread per (edge, float4-quad): gather h[src], atomic scatter-add into agg[dst]
template <int F>
__global__ void scatter_add(const float* __restrict__ h, const int* __restrict__ src,
                            const int* __restrict__ dst, float* __restrict__ agg, int E) {
    constexpr int QUADS = F / 4;
    long long t = (long long)blockIdx.x * blockDim.x + threadIdx.x;
    if (t >= (long long)E * QUADS) return;
    int e = (int)(t / QUADS);
    int q = (int)(t % QUADS);
    int s = src[e];
    int d = dst[e];
    const float4 v = *(const float4*)(h + (long long)s * F + 4 * q);
    float* o = agg + (long long)d * F + 4 * q;
    atomicAdd(o + 0, v.x);
    atomicAdd(o + 1, v.y);
    atomicAdd(o + 2, v.z);
    atomicAdd(o + 3, v.w);
}

// ---------------- WMMA transform ----------------
// One wave computes a 16-node row tile of:
//   out = sigmoid?( (agg*inv) @ W_rel + h_in @ W_root + bias )
// with chained V_WMMA_F32_16X16X4_F32 (exact fp32, K steps of 4, fully
// unrolled via compile-time F_IN/F_OUT). Weight loads use clamped always-valid
// addresses + value select so no divergent branches enter the WMMA loop
// (EXEC stays all-1s as the ISA requires).
template <int F_IN, int F_OUT, bool HAS_REL>
__global__ void gcn_wmma(const float* __restrict__ h_in,   // [n, F_IN]
                         const float* __restrict__ agg,    // [n, F_IN]   (HAS_REL)
                         const float* __restrict__ inv,    // [n]         (HAS_REL)
                         const float* __restrict__ W_rel,  // [F_IN,F_OUT](HAS_REL)
                         const float* __restrict__ bias,   // [F_OUT]
                         const float* __restrict__ W_root, // [F_IN, F_OUT]
                         float* __restrict__ out,          // [n, F_OUT]
                         int n, int apply_sig) {
    const int lane = threadIdx.x & 31;
    const int wv   = threadIdx.x >> 5;
    const int tile = blockIdx.x * (blockDim.x >> 5) + wv;
    const int row0 = tile * 16;
    if (row0 >= n) return;                 // wave-uniform exit

    const int col  = lane & 15;            // A-operand row / B-C-D column
    const int half = lane >> 4;
    int arow = row0 + col;
    if (arow >= n) arow = n - 1;           // per-lane clamp (select, no branch)
    const float iv = HAS_REL ? inv[arow] : 0.0f;

#pragma unroll
    for (int cb = 0; cb < F_OUT; cb += 16) {
        v8f c = {0.f, 0.f, 0.f, 0.f, 0.f, 0.f, 0.f, 0.f};
        const int  c0     = cb + col;
        const bool cvalid = (c0 < F_OUT);
        const int  cc     = cvalid ? c0 : (F_OUT - 1);   // clamped, always in-range

#pragma unroll
        for (int k = 0; k < F_IN; k += 4) {
            const int k0 = k + 2 * half;   // ISA 16x4 fp32 A layout: VGPR0=K{0,2}, VGPR1=K{1,3}

            // root term: h_in @ W_root
            v2f a, b;
            a.x = h_in[(long long)arow * F_IN + k0];
            a.y = h_in[(long long)arow * F_IN + k0 + 1];
            const float wr0 = W_root[k0 * F_OUT + cc];
            const float wr1 = W_root[(k0 + 1) * F_OUT + cc];
            b.x = cvalid ? wr0 : 0.0f;
            b.y = cvalid ? wr1 : 0.0f;
            c = __builtin_amdgcn_wmma_f32_16x16x4_f32(false, a, false, b,
                                                      (short)0, c, false, false);

            // relation term: (agg * inv) @ W_rel
            if (HAS_REL) {
                v2f am, bm;
                am.x = agg[(long long)arow * F_IN + k0] * iv;
                am.y = agg[(long long)arow * F_IN + k0 + 1] * iv;
                const float wm0 = W_rel[k0 * F_OUT + cc];
                const float wm1 = W_rel[(k0 + 1) * F_OUT + cc];
                bm.x = cvalid ? wm0 : 0.0f;
                bm.y = cvalid ? wm1 : 0.0f;
                c = __builtin_amdgcn_wmma_f32_16x16x4_f32(false, am, false, bm,
                                                          (short)0, c, false, false);
            }
        }

        if (cvalid) {
            const float bv = bias[c0];
#pragma unroll
            for (int r = 0; r < 8; ++r) {
                const int orow = row0 + r + 8 * half;   // C/D layout: VGPR r -> row r + 8*half
                if (orow < n) {
                    float v = c[r] + bv;
                    if (apply_sig) v = 1.0f / (1.0f + expf(-v));
                    out[(long long)orow * F_OUT + c0] = v;
                }
            }
        }
    }
}

// final 32 -> 1 projection: plain per-node dot (WMMA would waste 15/16 of the tile)
__global__ void fc2_kernel(const float* __restrict__ h, const float* __restrict__ W,
                           const float* __restrict__ b, float* __restrict__ out, int n) {
    int i = blockIdx.x * blockDim.x + threadIdx.x;
    if (i >= n) return;
    float acc = b[0];
#pragma unroll
    for (int j = 0; j < 32; ++j) acc += h[(long long)i * 32 + j] * W[j];
    out[i] = acc;
}

// ---------------- launch ----------------

static inline int cdiv_ll(long long a, long long b) { return (int)((a + b - 1) / b); }

extern "C" void kernel_launch(void* const* d_in, const int* in_sizes, int n_in,
                              void* d_out, int out_size, void* d_ws, size_t ws_size,
                              hipStream_t stream) {
    const float* x       = (const float*)d_in[0];
    const int*   ei      = (const int*)  d_in[1];
    const float* W_rel1  = (const float*)d_in[2];
    const float* b1      = (const float*)d_in[3];
    const float* W_root1 = (const float*)d_in[4];
    const float* W_rel2  = (const float*)d_in[5];
    const float* b2      = (const float*)d_in[6];
    const float* W_root2 = (const float*)d_in[7];
    const float* W_rel3  = (const float*)d_in[8];
    const float* b3      = (const float*)d_in[9];
    const float* W_root3 = (const float*)d_in[10];
    const float* fc1_W   = (const float*)d_in[11];
    const float* fc1_b   = (const float*)d_in[12];
    const float* fc2_W   = (const float*)d_in[13];
    const float* fc2_b   = (const float*)d_in[14];

    const int N = in_sizes[0] / 16;   // x is [N, 16]
    const int E = in_sizes[1] / 2;    // edge_index is [2, E]
    const int* src = ei;
    const int* dst = ei + E;

    // workspace layout (floats)
    float*    ws  = (float*)d_ws;
    unsigned* cnt = (unsigned*)ws;                 // N u32
    float*    inv = ws + N;                        // N
    float*    agg = ws + 2LL * N;                  // N*16 (reused each layer)
    float*    h1  = agg + (long long)N * 16;       // N*8
    float*    h2  = h1  + (long long)N * 8;        // N*16
    float*    h3  = h2  + (long long)N * 16;       // N*8
    float*    h4  = h3  + (long long)N * 8;        // N*32

    const int TB = 256;
    const int tiles = cdiv_ll(N, 16);
    const int gcn_blocks = cdiv_ll(tiles, TB / 32);

    // degree / inverse degree (computed once, reused by all three conv layers)
    zero_f32<<<cdiv_ll(N, TB), TB, 0, stream>>>((float*)cnt, N);
    degree_count<<<cdiv_ll(E, TB), TB, 0, stream>>>(dst, cnt, E);
    inv_degree<<<cdiv_ll(N, TB), TB, 0, stream>>>(cnt, inv, N);

    // layer 1: 16 -> 8
    zero_f32<<<cdiv_ll((long long)N * 16, TB), TB, 0, stream>>>(agg, (long long)N * 16);
    scatter_add<16><<<cdiv_ll((long long)E * 4, TB), TB, 0, stream>>>(x, src, dst, agg, E);
    gcn_wmma<16, 8, true><<<gcn_blocks, TB, 0, stream>>>(x, agg, inv, W_rel1, b1, W_root1, h1, N, 1);

    // layer 2: 8 -> 16
    zero_f32<<<cdiv_ll((long long)N * 8, TB), TB, 0, stream>>>(agg, (long long)N * 8);
    scatter_add<8><<<cdiv_ll((long long)E * 2, TB), TB, 0, stream>>>(h1, src, dst, agg, E);
    gcn_wmma<8, 16, true><<<gcn_blocks, TB, 0, stream>>>(h1, agg, inv, W_rel2, b2, W_root2, h2, N, 1);

    // layer 3: 16 -> 8
    zero_f32<<<cdiv_ll((long long)N * 16, TB), TB, 0, stream>>>(agg, (long long)N * 16);
    scatter_add<16><<<cdiv_ll((long long)E * 4, TB), TB, 0, stream>>>(h2, src, dst, agg, E);
    gcn_wmma<16, 8, true><<<gcn_blocks, TB, 0, stream>>>(h2, agg, inv, W_rel3, b3, W_root3, h3, N, 1);

    // fc1: 8 -> 32 (root-only GEMM path, two statically-unrolled 16-column tiles)
    gcn_wmma<8, 32, false><<<gcn_blocks, TB, 0, stream>>>(h3, nullptr, nullptr, nullptr, fc1_b, fc1_W, h4, N, 1);

    // fc2: 32 -> 1
    fc2_kernel<<<cdiv_ll(N, TB), TB, 0, stream>>>(h4, fc2_W, fc2_b, (float*)d_out, N);
}